// SingleStepRecurrentAttentionCell_10419590660436
// MI455X (gfx1250) — compile-verified
//
#include <hip/hip_runtime.h>

typedef __attribute__((ext_vector_type(2))) float v2f;
typedef __attribute__((ext_vector_type(8))) float v8f;

#define BB 64
#define HH 16
#define TT 4
#define DK 128
#define DV 256

// One workgroup per (b,h). 512 threads = 16 waves (wave32).
// Wave w owns v-columns [16w, 16w+16). State kept in registers in the
// V_WMMA C/D 16x16 f32 layout: tile j (k-rows 16j..16j+15), VGPR m holds
// S[16j + m + 8*(lane>=16)][16w + lane%16].
__global__ __launch_bounds__(512)
void gdn_recurrent_kernel(const float* __restrict__ q,
                          const float* __restrict__ k,
                          const float* __restrict__ v,
                          const float* __restrict__ g,
                          const float* __restrict__ beta,
                          const float* __restrict__ s0,
                          float* __restrict__ out)
{
    __shared__ float sq[TT * DK];
    __shared__ float sk[TT * DK];
    __shared__ float sv[TT * DV];
    __shared__ float sg[TT];
    __shared__ float sb[TT];

    const int bh   = blockIdx.x;     // 0..B*H-1
    const int tid  = threadIdx.x;    // 0..511
    const int wave = tid >> 5;       // 0..15
    const int lane = tid & 31;       // 0..31
    const int lo16 = lane & 15;
    const int hi   = lane >> 4;      // 0 (rows m) or 1 (rows m+8)

    // ---- stage per-head inputs into LDS ----
    const size_t qkBase = (size_t)bh * TT * DK;
    const size_t vBase  = (size_t)bh * TT * DV;
    for (int i = tid; i < TT * DK; i += 512) { sq[i] = q[qkBase + i]; sk[i] = k[qkBase + i]; }
    for (int i = tid; i < TT * DV; i += 512) { sv[i] = v[vBase + i]; }
    if (tid < TT) { sg[tid] = g[(size_t)bh * TT + tid]; sb[tid] = beta[(size_t)bh * TT + tid]; }
    __syncthreads();

    // ---- load state once into registers (WMMA C/D layout), non-temporal ----
    const int vcol = (wave << 4) + lo16;        // this lane's column
    const size_t stBase = (size_t)bh * DK * DV; // state offset for this head
    v8f S[8];
    #pragma unroll
    for (int j = 0; j < 8; ++j) {
        #pragma unroll
        for (int m = 0; m < 8; ++m) {
            const int kr = (j << 4) + m + (hi << 3);
            S[j][m] = __builtin_nontemporal_load(&s0[stBase + (size_t)kr * DV + vcol]);
        }
    }

    // ---- T recurrent steps, state resident in registers ----
    #pragma unroll
    for (int t = 0; t < TT; ++t) {
        const float decay = __expf(sg[t]);
        const float bt    = sb[t];
        const float vt    = sv[t * DV + vcol];

        // 1) decay:  S *= exp(g_t)
        #pragma unroll
        for (int j = 0; j < 8; ++j)
            #pragma unroll
            for (int m = 0; m < 8; ++m)
                S[j][m] *= decay;

        // 2) kv_mem[v] = sum_k S[k][v] * k_t[k]; vectorized LDS reads (b128)
        float part = 0.f;
        #pragma unroll
        for (int j = 0; j < 8; ++j) {
            const float* kp = &sk[t * DK + (j << 4) + (hi << 3)];
            const float4 a = *(const float4*)(kp);
            const float4 b = *(const float4*)(kp + 4);
            part += S[j][0] * a.x + S[j][1] * a.y + S[j][2] * a.z + S[j][3] * a.w
                  + S[j][4] * b.x + S[j][5] * b.y + S[j][6] * b.z + S[j][7] * b.w;
        }
        const float kv = part + __shfl_xor(part, 16, 32); // combine row-halves

        // 3) delta[v] = (v_t[v] - kv_mem[v]) * beta_t
        const float delta = (vt - kv) * bt;

        // 4) rank-1 update S += k_t (x) delta via V_WMMA_F32_16X16X4_F32.
        //    A (16x4): col K=0 = k chunk (lanes 0-15), K=1..3 = 0.
        //    B (4x16): row K=0 = delta chunk (lanes 0-15), rows 1..3 = 0.
        //    Loads hoisted & unconditional -> no exec-masked ds_load; mask via cndmask.
        float ka[8];
        #pragma unroll
        for (int j = 0; j < 8; ++j) ka[j] = sk[t * DK + (j << 4) + lo16];

        v2f Bop;
        Bop[0] = hi ? 0.f : delta;
        Bop[1] = 0.f;
        #pragma unroll
        for (int j = 0; j < 8; ++j) {
            v2f Aop;
            Aop[0] = hi ? 0.f : ka[j];
            Aop[1] = 0.f;
            S[j] = __builtin_amdgcn_wmma_f32_16x16x4_f32(
                false, Aop, false, Bop, (short)0, S[j], false, false);
        }

        // 5) out[v] = sum_k S[k][v] * q_t[k]
        float opart = 0.f;
        #pragma unroll
        for (int j = 0; j < 8; ++j) {
            const float* qp = &sq[t * DK + (j << 4) + (hi << 3)];
            const float4 a = *(const float4*)(qp);
            const float4 b = *(const float4*)(qp + 4);
            opart += S[j][0] * a.x + S[j][1] * a.y + S[j][2] * a.z + S[j][3] * a.w
                   + S[j][4] * b.x + S[j][5] * b.y + S[j][6] * b.z + S[j][7] * b.w;
        }
        const float osum = opart + __shfl_xor(opart, 16, 32);
        if (hi == 0)
            out[(size_t)bh * TT * DV + (size_t)t * DV + vcol] = osum;
    }

    // ---- write final state (second region of d_out), non-temporal ----
    const size_t stateOut = (size_t)BB * HH * TT * DV + stBase;
    #pragma unroll
    for (int j = 0; j < 8; ++j) {
        #pragma unroll
        for (int m = 0; m < 8; ++m) {
            const int kr = (j << 4) + m + (hi << 3);
            __builtin_nontemporal_store(S[j][m], &out[stateOut + (size_t)kr * DV + vcol]);
        }
    }
}

extern "C" void kernel_launch(void* const* d_in, const int* in_sizes, int n_in,
                              void* d_out, int out_size, void* d_ws, size_t ws_size,
                              hipStream_t stream) {
    const float* q    = (const float*)d_in[0];
    const float* k    = (const float*)d_in[1];
    const float* v    = (const float*)d_in[2];
    const float* g    = (const float*)d_in[3];
    const float* beta = (const float*)d_in[4];
    const float* s0   = (const float*)d_in[5];
    float* out = (float*)d_out;

    dim3 grid(BB * HH);   // 1024 workgroups, one per (b,h)
    dim3 block(512);      // 16 waves of 32
    gdn_recurrent_kernel<<<grid, block, 0, stream>>>(q, k, v, g, beta, s0, out);
}